// WindowAttention3D_tf_1030792151224
// MI455X (gfx1250) — compile-verified
//
#include <hip/hip_runtime.h>

// ---------------- problem constants ----------------
#define DIMC   128
#define HEADS  4
#define HD     32
#define NTOK   98
#define NPAD   112          // 7 * 16
#define NWIN   64
#define NBATCH 2048
#define QKSCALE 0.17677669529663687f  // 32^-0.5

// ---------------- LDS layout (floats) ----------------
#define QKV_STRIDE 36       // 32 + 4 pad
#define ATT_STRIDE 116      // 112 + 4 pad
#define OUT_STRIDE 132      // 128 + 4 pad (TDM pad: 128dw interval + 4dw)
#define PW_STRIDE  136      // 128 + 8 pad (TDM pad: 128dw interval + 8dw)
#define WCH_STRIDE 48       // staged qkv_w chunk: 48 cols, no pad

#define QKV_FLTS  (3 * HEADS * NPAD * QKV_STRIDE)   // 48384
#define ATT_OFF   QKV_FLTS
#define ATT_FLTS  (NPAD * ATT_STRIDE)               // 12992
#define XS_OFF    (ATT_OFF + ATT_FLTS)              // 61376
#define XS_FLTS   (NPAD * OUT_STRIDE)               // 14784
#define SMEM_FLTS (XS_OFF + XS_FLTS)                // 76160 floats = 304640 B (< 320KB WGP LDS)

#define WCH_BUF1  6400      // second W staging buffer offset inside attn region

typedef __attribute__((ext_vector_type(2))) float v2f;
typedef __attribute__((ext_vector_type(8))) float v8f;
typedef __attribute__((ext_vector_type(4))) unsigned int v4u;
typedef __attribute__((ext_vector_type(8))) int v8i;
typedef __attribute__((ext_vector_type(4))) int v4i;

// Explicit LDS (address_space(3)) element type: forces ds_load/ds_store
// codegen (no FLAT fallback when pointer provenance gets murky).
typedef __attribute__((address_space(3))) float lds_f32;

// ---------------------------------------------------------------
// TDM: DMA a 2D f32 tile (width x rows, row stride in elements) from
// global to LDS, optionally inserting pad_amount dwords into the LDS
// stream every 2^(pad_interval+1) dwords. Issued once per calling wave.
// Descriptor packing per CDNA5 ISA 8.3/8.4 (D# groups 0/1; 2D -> extra
// groups zero). Tracked by TENSORcnt. 6-arg builtin form (clang-23).
// ---------------------------------------------------------------
__device__ __forceinline__ void tdm_load_2d_f32(const float* gsrc, unsigned lds_byte_addr,
                                                unsigned width, unsigned rows,
                                                unsigned row_stride,
                                                unsigned pad_interval_code,
                                                unsigned pad_amount_code,
                                                bool pad_en) {
  unsigned long long ga = (unsigned long long)(const void*)gsrc;
  v4u g0;
  g0.x = 1u;                                            // count=1 (valid), user mode
  g0.y = lds_byte_addr;                                 // lds_addr [63:32]
  g0.z = (unsigned)(ga & 0xFFFFFFFFu);                  // global_addr [95:64]
  g0.w = (unsigned)((ga >> 32) & 0x01FFFFFFu) | (2u << 30); // global_addr hi | type=2
  unsigned w0 = (2u << 16);                             // data_size = 4 bytes
  if (pad_en) w0 |= (1u << 20) | (pad_interval_code << 22) | (pad_amount_code << 25);
  v8i g1;
  g1[0] = (int)w0;                                      // mask=0, data_size, pad cfg
  g1[1] = (int)((width & 0xFFFFu) << 16);               // tensor_dim0[15:0]
  g1[2] = (int)(((width >> 16) & 0xFFFFu) | ((rows & 0xFFFFu) << 16)); // dim0 hi | dim1 lo
  g1[3] = (int)(((rows >> 16) & 0xFFFFu) | ((width & 0xFFFFu) << 16)); // dim1 hi | tile_dim0
  g1[4] = (int)(rows & 0xFFFFu);                        // tile_dim1 (tile_dim2 = 0)
  g1[5] = (int)row_stride;                              // tensor_dim0_stride[31:0]
  g1[6] = 0;                                            // stride hi | dim1_stride lo
  g1[7] = 0;
  v4i gz4 = {0, 0, 0, 0};
  v8i gz8 = {0, 0, 0, 0, 0, 0, 0, 0};
  __builtin_amdgcn_tensor_load_to_lds(g0, g1, gz4, gz4, gz8, 0);
}

__device__ __forceinline__ unsigned lds_addr_of(const lds_f32* p) {
  return (unsigned)(unsigned long long)p;   // ptrtoint of AS(3) = LDS byte offset
}

// ---------------------------------------------------------------
// f32 WMMA 16x16 tile: C += A[16xK] * B[KxN], A row-major (lda),
// B row-major (ldb), tile pointers pre-offset to (row 0, col 0).
// A/B live in LDS (AS3) -> pure ds_load operand fetch.
// ---------------------------------------------------------------
__device__ __forceinline__ v8f wmma_AB(const lds_f32* A, int lda,
                                       const lds_f32* B, int ldb,
                                       int K, v8f acc, int lane) {
  const int m  = lane & 15;
  const int ko = (lane >> 4) << 1;   // 0 or 2
  for (int k = 0; k < K; k += 4) {
    v2f a, b;
    a.x = A[m * lda + k + ko];
    a.y = A[m * lda + k + ko + 1];
    b.x = B[(k + ko) * ldb + m];
    b.y = B[(k + ko + 1) * ldb + m];
    acc = __builtin_amdgcn_wmma_f32_16x16x4_f32(false, a, false, b,
                                                (short)0, acc, false, false);
  }
  return acc;
}

// B supplied transposed: Bt is [N][K] row-major (ldbt) -> computes A * Bt^T
__device__ __forceinline__ v8f wmma_ABt(const lds_f32* A, int lda,
                                        const lds_f32* Bt, int ldbt,
                                        int K, v8f acc, int lane) {
  const int m  = lane & 15;
  const int ko = (lane >> 4) << 1;
  for (int k = 0; k < K; k += 4) {
    v2f a, b;
    a.x = A[m * lda + k + ko];
    a.y = A[m * lda + k + ko + 1];
    b.x = Bt[m * ldbt + k + ko];
    b.y = Bt[m * ldbt + k + ko + 1];
    acc = __builtin_amdgcn_wmma_f32_16x16x4_f32(false, a, false, b,
                                                (short)0, acc, false, false);
  }
  return acc;
}

// ---------------------------------------------------------------
// One workgroup (8 wave32) handles one window b of 2048.
// ---------------------------------------------------------------
__global__ __launch_bounds__(256)
void win_attn3d_kernel(const float* __restrict__ x,
                       const float* __restrict__ mask,
                       const float* __restrict__ qkv_w,
                       const float* __restrict__ qkv_b,
                       const float* __restrict__ table,
                       const int*   __restrict__ rel_index,
                       const float* __restrict__ proj_w,
                       const float* __restrict__ proj_b,
                       float* __restrict__ out) {
  extern __shared__ float smem_generic[];
  lds_f32* smem   = (lds_f32*)smem_generic;   // explicit AS(3) view
  lds_f32* qkvBuf = smem;                     // [3][4][112][36]  q/k/v per head
  lds_f32* attn   = smem + ATT_OFF;           // [112][116] scores / W staging
  lds_f32* xs     = smem + XS_OFF;            // [112][132] x tile, later out accum
  lds_f32* pw     = smem;                     // Phase D: proj_w staged (aliases qkvBuf)

  const int b    = blockIdx.x;
  const int w    = b & (NWIN - 1);
  const int tid  = threadIdx.x;
  const int lane = tid & 31;
  const int wid  = tid >> 5;                 // wave id (wave32!)
  const int c    = lane & 15;                // tile column held by this lane
  const int rb   = (lane >> 4) << 3;         // C/D row base: 0 or 8

  const float* xg = x + (size_t)b * NTOK * DIMC;

  // ---- Phase A: TDM-stage x[b] (98x128 -> stride-132 LDS) + first W chunk ----
  if (wid == 0) {
    tdm_load_2d_f32(xg, lds_addr_of(xs), DIMC, NTOK, DIMC,
                    /*interval: 128dw*/6, /*amount: 4dw*/3, true);
    tdm_load_2d_f32(qkv_w, lds_addr_of(attn), WCH_STRIDE, DIMC, 3 * DIMC,
                    0, 0, false);
  }
  for (int i = tid; i < (NPAD - NTOK) * OUT_STRIDE; i += 256)   // zero pad rows
    xs[(NTOK + i / OUT_STRIDE) * OUT_STRIDE + (i % OUT_STRIDE)] = 0.0f;
  if (wid == 0) __builtin_amdgcn_s_wait_tensorcnt(0);
  __syncthreads();

  // ---- Phase B: qkv = xs @ qkv_w + qkv_b (q pre-scaled), rows>=98 zeroed ----
  // 8 chunks of 48 columns; chunk ch+1 DMA overlaps chunk ch compute.
  // Double buffer selected by integer offset (keeps AS3 provenance).
  for (int ch = 0; ch < 8; ++ch) {
    if (wid == 0) __builtin_amdgcn_s_wait_tensorcnt(0);   // chunk ch landed
    __syncthreads();                                      // + chunk ch-1 consumed
    if (wid == 0 && ch + 1 < 8)
      tdm_load_2d_f32(qkv_w + (ch + 1) * WCH_STRIDE,
                      lds_addr_of(attn) + (unsigned)(((ch + 1) & 1) * WCH_BUF1 * 4),
                      WCH_STRIDE, DIMC, 3 * DIMC, 0, 0, false);
    const lds_f32* wb = attn + (ch & 1) * WCH_BUF1;
    for (int t = wid; t < 7 * 3; t += 8) {
      const int mt = t / 3, nt = t % 3;
      const int ct = ch * 3 + nt;            // global 16-col tile index (0..23)
      v8f acc = {};
      acc = wmma_AB(xs + mt * 16 * OUT_STRIDE, OUT_STRIDE,
                    wb + nt * 16, WCH_STRIDE, DIMC, acc, lane);
      const int s  = ct >> 3;                // 0=q 1=k 2=v
      const int h  = (ct >> 1) & 3;          // head
      const int db = (ct & 1) << 4;          // d base within head
      const float bias = qkv_b[ct * 16 + c];
      const float scl  = (s == 0) ? QKSCALE : 1.0f;
      lds_f32* dst = qkvBuf + ((s * HEADS + h) * NPAD) * QKV_STRIDE + db + c;
#pragma unroll
      for (int r = 0; r < 8; ++r) {
        const int row = mt * 16 + rb + r;
        float v = (acc[r] + bias) * scl;
        if (row >= NTOK) v = 0.0f;           // zero pad rows of q/k/v
        dst[row * QKV_STRIDE] = v;
      }
    }
  }
  __syncthreads();

  const float* maskw = mask + (size_t)w * NTOK * NTOK;

  // ---- Phase C: per-head attention ----
  for (int h = 0; h < HEADS; ++h) {
    const lds_f32* qh = qkvBuf + (0 * HEADS + h) * NPAD * QKV_STRIDE;
    const lds_f32* kh = qkvBuf + (1 * HEADS + h) * NPAD * QKV_STRIDE;
    const lds_f32* vh = qkvBuf + (2 * HEADS + h) * NPAD * QKV_STRIDE;

    // C1: S = q @ k^T, fused +rel_bias +window_mask; pad entries -> 0
    for (int t = wid; t < 49; t += 8) {
      const int mt = t / 7, nt = t % 7;
      v8f acc = {};
      acc = wmma_ABt(qh + mt * 16 * QKV_STRIDE, QKV_STRIDE,
                     kh + nt * 16 * QKV_STRIDE, QKV_STRIDE, HD, acc, lane);
      const int j = nt * 16 + c;
#pragma unroll
      for (int r = 0; r < 8; ++r) {
        const int i = mt * 16 + rb + r;
        float v = 0.0f;
        if (i < NTOK && j < NTOK) {
          const int ri = rel_index[i * NTOK + j];
          v = acc[r] + table[ri * HEADS + h] + maskw[i * NTOK + j];
        }
        attn[i * ATT_STRIDE + j] = v;
      }
    }
    __syncthreads();

    // C2: row softmax over the 98 valid columns (pad stays exactly 0)
    if (tid < NTOK) {
      lds_f32* row = attn + tid * ATT_STRIDE;
      float mx = -1e30f;
      for (int j = 0; j < NTOK; ++j) mx = fmaxf(mx, row[j]);
      float sum = 0.0f;
      for (int j = 0; j < NTOK; ++j) {
        const float e = __expf(row[j] - mx);
        row[j] = e;
        sum += e;
      }
      const float inv = 1.0f / sum;
      for (int j = 0; j < NTOK; ++j) row[j] *= inv;
    }
    __syncthreads();

    // C3: out[:, h*32 : h*32+32] = attn @ v_h   (K padded to 112, pads are 0)
    for (int t = wid; t < 14; t += 8) {
      const int mt = t / 2, nt = t % 2;
      v8f acc = {};
      acc = wmma_AB(attn + mt * 16 * ATT_STRIDE, ATT_STRIDE,
                    vh + nt * 16, QKV_STRIDE, NPAD, acc, lane);
      lds_f32* dst = xs + h * HD + nt * 16 + c;   // xs now serves as out accum
#pragma unroll
      for (int r = 0; r < 8; ++r)
        dst[(mt * 16 + rb + r) * OUT_STRIDE] = acc[r];
    }
    __syncthreads();
  }

  // ---- Phase D: final = out @ proj_w + proj_b -> global ----
  // qkv region is dead now; TDM-stage proj_w there with stride-136 padding.
  if (wid == 0) {
    tdm_load_2d_f32(proj_w, lds_addr_of(pw), DIMC, DIMC, DIMC,
                    /*interval: 128dw*/6, /*amount: 8dw*/7, true);
    __builtin_amdgcn_s_wait_tensorcnt(0);
  }
  __syncthreads();

  float* og = out + (size_t)b * NTOK * DIMC;
  for (int t = wid; t < 7 * 8; t += 8) {
    const int mt = t / 8, nt = t % 8;
    v8f acc = {};
    acc = wmma_AB(xs + mt * 16 * OUT_STRIDE, OUT_STRIDE,
                  pw + nt * 16, PW_STRIDE, DIMC, acc, lane);
    const float pb = proj_b[nt * 16 + c];
#pragma unroll
    for (int r = 0; r < 8; ++r) {
      const int row = mt * 16 + rb + r;
      if (row < NTOK)
        og[row * DIMC + nt * 16 + c] = acc[r] + pb;
    }
  }
}

// ---------------------------------------------------------------
extern "C" void kernel_launch(void* const* d_in, const int* in_sizes, int n_in,
                              void* d_out, int out_size, void* d_ws, size_t ws_size,
                              hipStream_t stream) {
  (void)in_sizes; (void)n_in; (void)d_ws; (void)ws_size; (void)out_size;
  const float* x      = (const float*)d_in[0];
  const float* mask   = (const float*)d_in[1];
  const float* qkv_w  = (const float*)d_in[2];
  const float* qkv_b  = (const float*)d_in[3];
  const float* table  = (const float*)d_in[4];
  const int*   relidx = (const int*)  d_in[5];
  const float* proj_w = (const float*)d_in[6];
  const float* proj_b = (const float*)d_in[7];
  float* outp = (float*)d_out;

  const size_t smem_bytes = (size_t)SMEM_FLTS * sizeof(float);
  win_attn3d_kernel<<<NBATCH, 256, smem_bytes, stream>>>(
      x, mask, qkv_w, qkv_b, table, relidx, proj_w, proj_b, outp);
}